// TextLSTM_26456998543692
// MI455X (gfx1250) — compile-verified
//
#include <hip/hip_runtime.h>
#include <cmath>

typedef float v2f __attribute__((ext_vector_type(2)));
typedef float v8f __attribute__((ext_vector_type(8)));

#define B_SZ 1024
#define S_SZ 512
#define E_SZ 50
#define H_SZ 50
#define G_SZ 200              // 4*H
#define K_SZ 100              // E + H (exactly 25 k-steps of 4)
#define NT 13                 // N tiles of 16 covering 208 (>=200)
#define ROWS 16               // batch rows per block (1 M tile) -> 64 blocks
#define THREADS (NT * 32)     // 416 threads = 13 waves
#define GSTRIDE 204           // padded gate-row stride (bank-conflict-free)

// CDNA5 hardware tanh (v_tanh_f32) if the builtin is available.
#if __has_builtin(__builtin_amdgcn_tanhf)
#define FAST_TANH(x) __builtin_amdgcn_tanhf(x)
#elif __has_builtin(__builtin_amdgcn_tanh_f32)
#define FAST_TANH(x) __builtin_amdgcn_tanh_f32(x)
#else
#define FAST_TANH(x) tanhf(x)
#endif

__device__ __forceinline__ float fast_sigmoid(float x) {
    // sigmoid(x) = 0.5 * tanh(x/2) + 0.5  -> single trans op + FMA
    return __builtin_fmaf(0.5f, FAST_TANH(0.5f * x), 0.5f);
}

__global__ __launch_bounds__(THREADS)
void lstm_fused_kernel(const float* __restrict__ X,
                       const float* __restrict__ Wih,
                       const float* __restrict__ Whh,
                       const float* __restrict__ bih,
                       const float* __restrict__ bhh,
                       float* __restrict__ out)
{
    __shared__ float Z[ROWS][K_SZ];     // A staging: [x_t | h_{t-1}]
    __shared__ float Gs[ROWS][GSTRIDE]; // activated gates (200 valid cols)
    __shared__ float Cs[ROWS][H_SZ];    // cell state

    const int tid  = threadIdx.x;
    const int wave = tid >> 5;
    const int lane = tid & 31;
    const int b0   = blockIdx.x * ROWS;

    // ---- per-lane matrix coordinates ----
    const int n     = wave * 16 + (lane & 15); // gate column owned by this lane
    const int khalf = (lane >> 4) * 2;         // K sub-offset (A and B frags)
    const int m0    = lane & 15;               // row within 16-row tile
    const int rbase = (lane >> 4) * 8;         // C-tile row base (v / v+8 split)

    // ---- preload W fragments (B-matrix layout) into registers: 25 x v2f ----
    v2f wfrag[25];
#pragma unroll
    for (int kk = 0; kk < 25; ++kk) {
        const int k = kk * 4 + khalf;          // even, k+1 <= 99
        float w0 = 0.f, w1 = 0.f;
        if (n < G_SZ) {
            w0 = (k     < E_SZ) ? Wih[n * E_SZ + k]     : Whh[n * H_SZ + (k - E_SZ)];
            w1 = (k + 1 < E_SZ) ? Wih[n * E_SZ + k + 1] : Whh[n * H_SZ + (k + 1 - E_SZ)];
        }
        wfrag[kk].x = w0;
        wfrag[kk].y = w1;
    }
    const float bias_n = (n < G_SZ) ? (bih[n] + bhh[n]) : 0.f;
    const bool  istanh = (n < G_SZ) && ((n / 50) == 2);  // gate order i,f,g,o

    // ---- per-thread x prefetch pointers (advance by E_SZ floats per step) ----
    // Thread covers flat indices tid and min(tid+THREADS, ROWS*E_SZ-1); the
    // clamped duplicate loads/stores the same value to the same slot (benign).
    const int pidx0 = tid;                                     // < 800 always
    const int pidx1 = (tid + THREADS < ROWS * E_SZ) ? tid + THREADS
                                                    : ROWS * E_SZ - 1;
    const int zr0 = pidx0 / E_SZ, ze0 = pidx0 % E_SZ;
    const int zr1 = pidx1 / E_SZ, ze1 = pidx1 % E_SZ;
    const float* xp0 = X + (size_t)(b0 + zr0) * S_SZ * E_SZ + ze0;
    const float* xp1 = X + (size_t)(b0 + zr1) * S_SZ * E_SZ + ze1;

    // ---- init: h = 0, c = 0, load x_0 ----
    for (int idx = tid; idx < ROWS * H_SZ; idx += THREADS) {
        const int r = idx / H_SZ, j = idx % H_SZ;
        Z[r][E_SZ + j] = 0.f;
        Cs[r][j]       = 0.f;
    }
    Z[zr0][ze0] = xp0[0];
    Z[zr1][ze1] = xp1[0];
    xp0 += E_SZ;   // now point at x_1
    xp1 += E_SZ;
    __syncthreads();

    float xr0 = 0.f, xr1 = 0.f;

    for (int s = 0; s < S_SZ; ++s) {
        // -- prefetch x_{s+1} (uniform guard; latency hides under WMMA loop) --
        if (s + 1 < S_SZ) {
            xr0 = xp0[0];
            xr1 = xp1[0];
            xp0 += E_SZ;
            xp1 += E_SZ;
        }

        // -- gates = [x|h] * Wc^T + bias : 25 k-steps on 4 independent chains --
        v8f accs[4];
#pragma unroll
        for (int v = 0; v < 8; ++v) {
            accs[0][v] = bias_n;
            accs[1][v] = 0.f;
            accs[2][v] = 0.f;
            accs[3][v] = 0.f;
        }

#pragma unroll
        for (int kk = 0; kk < 25; ++kk) {
            const int k = kk * 4 + khalf;
            v2f a;
            a.x = Z[m0][k];
            a.y = Z[m0][k + 1];
            accs[kk & 3] = __builtin_amdgcn_wmma_f32_16x16x4_f32(
                               false, a, false, wfrag[kk], (short)0,
                               accs[kk & 3], false, false);
        }

        // -- combine chains + per-lane activation, store to LDS --
        if (n < G_SZ) {
#pragma unroll
            for (int v = 0; v < 8; ++v) {
                const float g = (accs[0][v] + accs[1][v]) +
                                (accs[2][v] + accs[3][v]);
                Gs[rbase + v][n] = istanh ? FAST_TANH(g) : fast_sigmoid(g);
            }
        }
        __syncthreads();

        // -- commit prefetched x_{s+1}; elementwise c/h update --
        if (s + 1 < S_SZ) {
            Z[zr0][ze0] = xr0;
            Z[zr1][ze1] = xr1;
        }
        for (int idx = tid; idx < ROWS * H_SZ; idx += THREADS) {
            const int r = idx / H_SZ, j = idx % H_SZ;
            const float ig = Gs[r][j];
            const float fg = Gs[r][j + 50];
            const float gg = Gs[r][j + 100];
            const float og = Gs[r][j + 150];
            const float c  = fg * Cs[r][j] + ig * gg;
            Cs[r][j] = c;
            const float h = og * FAST_TANH(c);
            Z[r][E_SZ + j] = h;
            if (s == S_SZ - 1)
                out[(size_t)(b0 + r) * H_SZ + j] = h;
        }
        __syncthreads();
    }
}

extern "C" void kernel_launch(void* const* d_in, const int* in_sizes, int n_in,
                              void* d_out, int out_size, void* d_ws, size_t ws_size,
                              hipStream_t stream) {
    (void)in_sizes; (void)n_in; (void)d_ws; (void)ws_size; (void)out_size;
    const float* X   = (const float*)d_in[0];
    const float* Wih = (const float*)d_in[1];
    const float* Whh = (const float*)d_in[2];
    const float* bih = (const float*)d_in[3];
    const float* bhh = (const float*)d_in[4];
    float* out = (float*)d_out;

    dim3 grid(B_SZ / ROWS);   // 64 blocks (max parallelism the recurrence admits)
    dim3 block(THREADS);      // 416 threads = 13 wave32
    lstm_fused_kernel<<<grid, block, 0, stream>>>(X, Wih, Whh, bih, bhh, out);
}